// L2SchemaWeaver_2980707303819
// MI455X (gfx1250) — compile-verified
//
#include <hip/hip_runtime.h>

typedef float v2f __attribute__((ext_vector_type(2)));
typedef float v8f __attribute__((ext_vector_type(8)));

#define D_DIM   1024
#define COLS    8            // columns (d indices) per workgroup
#define BLOCK   512          // threads per block -> 16 waves
#define GROUPS  (BLOCK/COLS) // 64 row-groups
#define WAVES   (BLOCK/32)   // 16
#define KITERS  15
#define CLAMP_  1e-7f
#define EPS_    1e-8f

__device__ __forceinline__ float vsel8(v8f v, int i) {
    float r = v[0];
    r = (i == 1) ? v[1] : r;
    r = (i == 2) ? v[2] : r;
    r = (i == 3) ? v[3] : r;
    r = (i == 4) ? v[4] : r;
    r = (i == 5) ? v[5] : r;
    r = (i == 6) ? v[6] : r;
    r = (i == 7) ? v[7] : r;
    return r;
}

__global__ __launch_bounds__(BLOCK)
void karcher_mean_kernel(const float4* __restrict__ knots, float4* __restrict__ out, int N) {
    __shared__ float4 muS[COLS];
    __shared__ float  part[WAVES][COLS][3];

    const int tid  = threadIdx.x;
    const int col  = tid & (COLS - 1);
    const int grp  = tid / COLS;         // 0..GROUPS-1
    const int lane = tid & 31;
    const int wave = tid >> 5;
    const int d0   = blockIdx.x * COLS;  // first column of this block

    // mu = knots[0, d]  (row 0 of each column)
    if (tid < COLS) muS[tid] = knots[d0 + tid];
    __syncthreads();

    const float invN = 1.0f / (float)N;

    for (int it = 0; it < KITERS; ++it) {
        const float4 m4 = muS[col];
        const float mw = m4.x, mx = m4.y, my = m4.z, mz = m4.w;

        float sx = 0.0f, sy = 0.0f, sz = 0.0f;

        const float4* p = knots + (size_t)grp * D_DIM + (d0 + col);
        #pragma unroll 4
        for (int n = grp; n < N; n += GROUPS) {
            const float4 k = *p;
            p += (size_t)GROUPS * D_DIM;
            const float kw = k.x, kx = k.y, ky = k.z, kz = k.w;

            // rel = conj(mu) * k   (quaternion product, (w,x,y,z) order)
            const float rw = mw*kw + mx*kx + my*ky + mz*kz;
            const float rx = mw*kx - mx*kw - my*kz + mz*ky;
            const float ry = mw*ky + mx*kz - my*kw - mz*kx;
            const float rz = mw*kz - mx*ky + my*kx - mz*kw;

            // su2 log map (reference semantics: clip w, norm of raw xyz, eps guard)
            const float w  = fminf(fmaxf(rw, -1.0f + CLAMP_), 1.0f - CLAMP_);
            const float nv = sqrtf(rx*rx + ry*ry + rz*rz);
            const float th = acosf(w);
            const float sc = (nv > EPS_) ? (th / nv) : 1.0f;

            sx = __builtin_fmaf(rx, sc, sx);
            sy = __builtin_fmaf(ry, sc, sy);
            sz = __builtin_fmaf(rz, sc, sz);
        }

        // wave-level butterfly: lanes {L, L^8, L^16, L^24} share column L&7
        sx += __shfl_xor(sx, 16); sx += __shfl_xor(sx, 8);
        sy += __shfl_xor(sy, 16); sy += __shfl_xor(sy, 8);
        sz += __shfl_xor(sz, 16); sz += __shfl_xor(sz, 8);

        if (lane < COLS) {
            part[wave][lane][0] = sx;
            part[wave][lane][1] = sy;
            part[wave][lane][2] = sz;
        }
        __syncthreads();

        if (wave == 0) {
            // Cross-wave reduction of 16 per-wave partials via WMMA:
            // A(16x4) holds 4 wave-partials per chained call (documented A layout:
            // lane m -> K0,K1 ; lane m+16 -> K2,K3), B = ones (layout-invariant).
            // D[m][n] = sum_k A[m][k]  -> row m is the group-sum for column m.
            const int m  = lane & 15;
            const int hi = lane >> 4;          // 0: lanes 0-15, 1: lanes 16-31
            const v2f ones = {1.0f, 1.0f};
            v8f cx = {}, cy = {}, cz = {};

            #pragma unroll
            for (int chunk = 0; chunk < WAVES / 4; ++chunk) {
                const int g = chunk * 4 + hi * 2;     // this half-wave's two K slots
                v2f ax = {0.0f, 0.0f}, ay = {0.0f, 0.0f}, az = {0.0f, 0.0f};
                if (m < COLS) {
                    ax[0] = part[g    ][m][0]; ax[1] = part[g + 1][m][0];
                    ay[0] = part[g    ][m][1]; ay[1] = part[g + 1][m][1];
                    az[0] = part[g    ][m][2]; az[1] = part[g + 1][m][2];
                }
                cx = __builtin_amdgcn_wmma_f32_16x16x4_f32(false, ax, false, ones,
                                                           (short)0, cx, false, false);
                cy = __builtin_amdgcn_wmma_f32_16x16x4_f32(false, ay, false, ones,
                                                           (short)0, cy, false, false);
                cz = __builtin_amdgcn_wmma_f32_16x16x4_f32(false, az, false, ones,
                                                           (short)0, cz, false, false);
            }

            // C/D layout: lane n<16, component j -> D[j][n]. Leader lane L (<8)
            // handles column L and needs row L: value = c[L] on any lane < 16.
            if (lane < COLS) {
                const float vx = vsel8(cx, lane) * invN;
                const float vy = vsel8(cy, lane) * invN;
                const float vz = vsel8(cz, lane) * invN;

                // su2 exp map (reference semantics)
                const float th = sqrtf(vx*vx + vy*vy + vz*vz);
                const float s  = (th > EPS_) ? (sinf(th) / th) : 0.0f;
                const float dw = cosf(th);
                const float dx = vx * s, dy = vy * s, dz = vz * s;

                // mu_new = mu * delta, then normalize
                const float4 mm = muS[lane];
                const float aw = mm.x, ax_ = mm.y, ay_ = mm.z, az_ = mm.w;
                const float nw = aw*dw - ax_*dx - ay_*dy - az_*dz;
                const float nx = aw*dx + ax_*dw + ay_*dz - az_*dy;
                const float ny = aw*dy - ax_*dz + ay_*dw + az_*dx;
                const float nz = aw*dz + ax_*dy - ay_*dx + az_*dw;
                const float inv = 1.0f / sqrtf(nw*nw + nx*nx + ny*ny + nz*nz);
                muS[lane] = make_float4(nw*inv, nx*inv, ny*inv, nz*inv);
            }
        }
        __syncthreads();
    }

    if (tid < COLS) out[d0 + tid] = muS[tid];
}

extern "C" void kernel_launch(void* const* d_in, const int* in_sizes, int n_in,
                              void* d_out, int out_size, void* d_ws, size_t ws_size,
                              hipStream_t stream) {
    (void)n_in; (void)out_size; (void)d_ws; (void)ws_size;
    const float4* knots = (const float4*)d_in[0];
    float4* out = (float4*)d_out;
    const int N = in_sizes[0] / (D_DIM * 4);   // 8192
    dim3 grid(D_DIM / COLS);                   // 128 blocks
    dim3 block(BLOCK);                         // 512 threads (16 waves)
    karcher_mean_kernel<<<grid, block, 0, stream>>>(knots, out, N);
}